// NCAM3D_40827959116385
// MI455X (gfx1250) — compile-verified
//
#include <hip/hip_runtime.h>
#include <math.h>

// Problem constants (fixed by the reference's setup_inputs).
static constexpr int C_TOT   = 480;   // C1*C2
static constexpr int HW      = 225;   // 15*15
static constexpr int TILES   = 240;   // owned tiles per block (half of C)
static constexpr int HALO    = 5;     // extra halo tiles whose means we need
static constexpr int THREADS = 1024;  // 32 waves (wave32)

#if defined(__has_builtin)
#  if __has_builtin(__builtin_amdgcn_global_load_async_to_lds_b32) && \
      __has_builtin(__builtin_amdgcn_s_wait_asynccnt)
#    define NCAM_USE_ASYNC 1
#  endif
#endif

#ifdef NCAM_USE_ASYNC
typedef __attribute__((address_space(1))) int g_int_t;   // global-AS int
typedef __attribute__((address_space(3))) int l_int_t;   // LDS-AS int
#endif

__global__ __launch_bounds__(THREADS) void ncam_fused_kernel(
    const float* __restrict__ x,
    const float* __restrict__ w1, const float* __restrict__ b1,
    const float* __restrict__ w2, const float* __restrict__ b2,
    float* __restrict__ out)
{
    __shared__ float s_meanF[C_TOT];   // full 15x15 mean, indexed by channel
    __shared__ float s_meanC[C_TOT];   // center 3x3 sum / 225, indexed by channel
    __shared__ float s_at[C_TOT];      // per-channel gate (owned channels only)
#ifdef NCAM_USE_ASYNC
    __shared__ __align__(16) float s_stage[THREADS / 32][240]; // 960B slot per wave
#endif

    const int tid  = threadIdx.x;
    const int wave = tid >> 5;
    const int lane = tid & 31;
    const int blk  = blockIdx.x;
    const int b    = blk >> 1;
    const int half = blk & 1;
    const size_t bbase = (size_t)b * C_TOT * HW;

    // ---------------- Phase 1: per-tile reductions (owned + halo) ----------
    for (int t = wave; t < TILES + HALO; t += THREADS / 32) {
        int c;
        if (t < TILES) {
            c = (half == 0) ? ((t < 120) ? t : t + 240) : (120 + t);
        } else {
            int i = t - TILES;
            // half 0 halo: {120,121,357,358,359}; half 1 halo: {117,118,119,360,361}
            c = (half == 0) ? ((i < 2) ? 120 + i : 355 + i)
                            : ((i < 3) ? 117 + i : 357 + i);
        }
        const float* src = x + bbase + (size_t)c * HW;

        float v[8];
#ifdef NCAM_USE_ASYNC
        {
            float* slot = &s_stage[wave][0];
            #pragma unroll
            for (int k = 0; k < 7; ++k) {
                __builtin_amdgcn_global_load_async_to_lds_b32(
                    (g_int_t*)(src + lane + 32 * k),
                    (l_int_t*)(slot + lane + 32 * k),
                    0, 0);
            }
            if (lane == 0) {
                __builtin_amdgcn_global_load_async_to_lds_b32(
                    (g_int_t*)(src + 224),
                    (l_int_t*)(slot + 224),
                    0, 0);
            }
            __builtin_amdgcn_s_wait_asynccnt(0);
            #pragma unroll
            for (int k = 0; k < 7; ++k) v[k] = slot[lane + 32 * k];
            v[7] = (lane == 0) ? slot[224] : 0.0f;
        }
#else
        #pragma unroll
        for (int k = 0; k < 7; ++k) v[k] = src[lane + 32 * k];
        v[7] = (lane == 0) ? src[224] : 0.0f;
#endif

        float fsum = 0.0f, csum = 0.0f;
        #pragma unroll
        for (int k = 0; k < 8; ++k) {
            const int idx = (k < 7) ? (lane + 32 * k) : 224;  // for k==7 only lane0 has data
            const float val = v[k];
            fsum += val;
            const int h = idx / 15, w = idx % 15;
            if (h >= 6 && h <= 8 && w >= 6 && w <= 8) csum += val;
        }
        // wave32 butterfly reduction
        #pragma unroll
        for (int off = 16; off > 0; off >>= 1) {
            fsum += __shfl_xor(fsum, off, 32);
            csum += __shfl_xor(csum, off, 32);
        }
        if (lane == 0) {
            s_meanF[c] = fsum * (1.0f / HW);
            s_meanC[c] = csum * (1.0f / HW);
        }
    }
    __syncthreads();

    // ---------------- Phase 2: conv(2x5, pad W=2) + sigmoids -> gate -------
    if (tid < TILES) {
        const int c = (half == 0) ? ((tid < 120) ? tid : tid + 240) : (120 + tid);
        float s1 = b1[0], s2 = b2[0];
        #pragma unroll
        for (int kw = 0; kw < 5; ++kw) {
            const int j = c + kw - 2;                 // padded W coordinate
            if (j >= 0 && j < C_TOT) {
                // row0 = forward means, row1 = channel-reversed means
                s1 += w1[kw] * s_meanF[j] + w1[5 + kw] * s_meanF[C_TOT - 1 - j];
                s2 += w2[kw] * s_meanC[j] + w2[5 + kw] * s_meanC[C_TOT - 1 - j];
            }
        }
        const float a1 = 1.0f / (1.0f + expf(-s1));
        const float a2 = 1.0f / (1.0f + expf(-s2));
        const float a  = a1 * a2;
        s_at[c] = 1.0f / (1.0f + expf(-2.0f * (a - 0.2f)));
    }
    __syncthreads();

    // ---------------- Phase 3: float4 scale of owned tiles -----------------
    // Owned data = 1 or 2 contiguous channel ranges; 54000 floats = 13500 vec4.
    const int nvec = (TILES * HW) / 4;   // 13500
    for (int v4 = tid; v4 < nvec; v4 += THREADS) {
        int startC, vv = v4;
        if (half == 0) {
            if (v4 < nvec / 2) { startC = 0; }
            else               { startC = 360; vv = v4 - nvec / 2; }
        } else {
            startC = 120;
        }
        const int e    = vv * 4;          // element offset within the range
        const int cLoc = e / HW;
        const int r    = e - cLoc * HW;
        const int c    = startC + cLoc;
        const size_t g = bbase + (size_t)startC * HW + e;   // 16B aligned

        const float4 xv = *reinterpret_cast<const float4*>(x + g);
        const float a0 = s_at[c];
        float4 ov;
        if (r <= HW - 4) {
            ov.x = xv.x * a0; ov.y = xv.y * a0; ov.z = xv.z * a0; ov.w = xv.w * a0;
        } else {
            // straddles tile boundary (r in {222,223,224}); c+1 is in-range here
            const float a1q = s_at[c + 1];
            ov.x = xv.x * ((r + 0 < HW) ? a0 : a1q);
            ov.y = xv.y * ((r + 1 < HW) ? a0 : a1q);
            ov.z = xv.z * ((r + 2 < HW) ? a0 : a1q);
            ov.w = xv.w * ((r + 3 < HW) ? a0 : a1q);
        }
        *reinterpret_cast<float4*>(out + g) = ov;
    }
}

extern "C" void kernel_launch(void* const* d_in, const int* in_sizes, int n_in,
                              void* d_out, int out_size, void* d_ws, size_t ws_size,
                              hipStream_t stream) {
    const float* x  = (const float*)d_in[0];
    const float* w1 = (const float*)d_in[1];
    const float* b1 = (const float*)d_in[2];
    const float* w2 = (const float*)d_in[3];
    const float* b2 = (const float*)d_in[4];
    float* out = (float*)d_out;

    const int B = in_sizes[0] / (C_TOT * HW);   // 512 for the reference setup
    dim3 grid(2 * B), block(THREADS);
    ncam_fused_kernel<<<grid, block, 0, stream>>>(x, w1, b1, w2, b2, out);
    (void)n_in; (void)out_size; (void)d_ws; (void)ws_size;
}